// Conv2_5D_depth_5669356836236
// MI455X (gfx1250) — compile-verified
//
#include <hip/hip_runtime.h>

// ---------------------------------------------------------------------------
// Conv2.5D depth-gated 3x3 conv for MI455X (gfx1250, wave32, WMMA).
//
// out[n,o,h,w] = sum_b sum_{c,kh,kw} W_b[o,c,kh,kw] * x[n,c,h+kh-1,w+kw-1]
//                                      * mask_b[n,(kh,kw),h,w]
//
// The depth mask depends only on (branch, tap, pixel) -- NOT on channel -- so
// it commutes with the channel reduction:
//   out[o,l] = sum_kk sum_b  m_b[kk,l] * ( W_b[:,:,kk] @ x_shift_kk )[o,l]
// Per tap: three mask-free K=64 sub-GEMMs (one per branch) over TWO adjacent
// 16-pixel N-subtiles -> 6 independent WMMA accumulator chains per wave
// (deep-pipelines the fp32 matrix unit) and 2x reuse of every A fragment
// (halves weight L2 traffic). Masks folded with FMAs per tap.
//
// LDS x tile is CHANNEL-PAIR INTERLEAVED: xs[((cp*3+r)*130+col)*2 + (c&1)],
// so each B fragment {x[c][j], x[c+1][j]} is one aligned 8-byte LDS load and
// both N-subtile fragments (128 B apart) merge into one ds_load_2addr_b64.
// Matrix op: V_WMMA_F32_16X16X4_F32 (exact fp32, D = A*B + C).
// Workgroup = 512 threads (16 waves) covers one image row: 64 O x 128 pixels.
// ---------------------------------------------------------------------------

typedef __attribute__((ext_vector_type(2))) float v2f;
typedef __attribute__((ext_vector_type(8))) float v8f;

#define KK    9     // 3x3 taps
#define CCH   64
#define OCH   64
#define HH    128
#define WW    128
#define LBLK  128   // pixels (full row) per workgroup
#define XCOLS 130   // LBLK + 2 halo
#define CPROW (3 * XCOLS * 2)   // floats per channel-pair block = 780

// Prepacked weights in d_ws:
//   wt[((b*KK + kk)*32 + (c>>1)) * 128 + o*2 + (c&1)] = W_b[o][c][kk]
// so a lane's A-fragment (two K-adjacent channels, one o) is one aligned v2f.
__global__ void prepack_weights(const float* __restrict__ w0,
                                const float* __restrict__ w1,
                                const float* __restrict__ w2,
                                float* __restrict__ wt) {
  int idx = blockIdx.x * blockDim.x + threadIdx.x;   // 3*9*64*64 = 110592
  int o  = idx & 63;
  int c  = (idx >> 6) & 63;
  int kk = (idx >> 12) % KK;
  int b  = idx / (KK * CCH * OCH);
  const float* w = (b == 0) ? w0 : (b == 1 ? w1 : w2);
  float v = w[(o * CCH + c) * KK + kk];
  wt[((size_t)((b * KK + kk) * 32 + (c >> 1))) * (OCH * 2) + o * 2 + (c & 1)] = v;
}

__global__ __launch_bounds__(512)
void conv25d_wmma_kernel(const float* __restrict__ x,
                         const float* __restrict__ depth,
                         const float* __restrict__ fx,
                         const float* __restrict__ wt,
                         float* __restrict__ out) {
  __shared__ float xs[(CCH / 2) * CPROW];   // 99840 B, channel-pair interleaved
  __shared__ float dsd[3 * XCOLS];          // depth rows h-1..h+1
  __shared__ float msk[3 * KK * LBLK];      // mask_b[kk][pixel] as 0.0/1.0

  const int tid = threadIdx.x;
  const int blk = blockIdx.x;            // 0..1023
  const int n   = blk >> 7;              // 128 rows per image
  const int h   = blk & 127;

  const float* xn = x + (size_t)n * CCH * HH * WW;
  const float* dn = depth + (size_t)n * HH * WW;
  const float inv_fx = 1.0f / fx[n];

  // ---- stage x tile (zero-padded halo), channel-pair interleaved ----
  // i = ((cp*3 + r)*XCOLS + col)*2 + e  maps to channel c = 2*cp + e.
  for (int i = tid; i < (CCH / 2) * CPROW; i += 512) {
    int e   = i & 1;
    int col = (i >> 1) % XCOLS;
    int r   = ((i >> 1) / XCOLS) % 3;
    int cp  = i / CPROW;
    int c   = cp * 2 + e;
    int gh = h - 1 + r;
    int gw = col - 1;
    float v = 0.0f;
    if ((unsigned)gh < HH && (unsigned)gw < WW)
      v = xn[((size_t)c * HH + gh) * WW + gw];
    xs[i] = v;
  }
  // ---- stage depth rows ----
  for (int i = tid; i < 3 * XCOLS; i += 512) {
    int col = i % XCOLS;
    int r   = i / XCOLS;
    int gh = h - 1 + r;
    int gw = col - 1;
    float v = 0.0f;
    if ((unsigned)gh < HH && (unsigned)gw < WW)
      v = dn[(size_t)gh * WW + gw];
    dsd[i] = v;
  }
  __syncthreads();

  // ---- depth gating masks: shared across all channels ----
  for (int i = tid; i < 3 * KK * LBLK; i += 512) {
    int j  = i & 127;           // pixel within row
    int kk = (i >> 7) % KK;
    int b  = i / (KK * LBLK);
    float center = dsd[XCOLS + (j + 1)];
    float grid   = center * inv_fx;          // PIXEL_SIZE = 1
    float halfg  = 0.5f * grid;
    int dh = kk / 3, dw = kk % 3;            // 0..2
    float dd  = dsd[dh * XCOLS + j + dw];    // col (j+1) + (dw-1)
    float ref = center + (float)(1 - b) * grid;  // b=0:+g  b=1:0  b=2:-g
    msk[i] = (fabsf(dd - ref) <= halfg) ? 1.0f : 0.0f;
  }
  __syncthreads();

  // ---- WMMA tiles: 16 waves cover 64(O) x 128(L); 16x32 tile per wave ----
  const int wave = tid >> 5;
  const int lane = tid & 31;
  const int lh   = lane >> 4;      // lane half selects K pair: K = 2*lh + comp
  const int jm   = lane & 15;      // A: M index (=o). B: N index (=pixel).
  const int o0   = (wave >> 2) * 16;
  const int j0   = (wave & 3) * 32;   // two N-subtiles: j0 and j0+16

  v8f acc0 = {}, acc1 = {};
  for (int kk = 0; kk < KK; ++kk) {
    const int dh = kk / 3, dw = kk % 3;
    // Per-lane (per output column) mask values, both subtiles, three branches.
    const float m0a = msk[(0 * KK + kk) * LBLK + j0 + jm];
    const float m1a = msk[(1 * KK + kk) * LBLK + j0 + jm];
    const float m2a = msk[(2 * KK + kk) * LBLK + j0 + jm];
    const float m0b = msk[(0 * KK + kk) * LBLK + j0 + 16 + jm];
    const float m1b = msk[(1 * KK + kk) * LBLK + j0 + 16 + jm];
    const float m2b = msk[(2 * KK + kk) * LBLK + j0 + 16 + jm];
    // A fragment bases: this lane's o column, channel-pair interleaved.
    const float* w0b = wt + (size_t)(0 * KK + kk) * 32 * (OCH * 2) + (size_t)(o0 + jm) * 2;
    const float* w1b = wt + (size_t)(1 * KK + kk) * 32 * (OCH * 2) + (size_t)(o0 + jm) * 2;
    const float* w2b = wt + (size_t)(2 * KK + kk) * 32 * (OCH * 2) + (size_t)(o0 + jm) * 2;
    // B fragment base for this lane's pixel column and this tap; per-cg step
    // advances by 2 channel-pair blocks (lane half lh selects odd/even pair).
    const float* xbase = xs + ((size_t)lh * 3 + dh) * (XCOLS * 2)
                            + (size_t)(j0 + jm + dw) * 2;

    // Six independent accumulator chains over K=64 (mask-free GEMMs);
    // each A fragment feeds two WMMAs (both N-subtiles); each B pair is one
    // 8-byte LDS load, both subtiles via one ds_load_2addr_b64 (128 B apart).
    v8f d0a = {}, d1a = {}, d2a = {}, d0b = {}, d1b = {}, d2b = {};
#pragma unroll
    for (int cg = 0; cg < 16; ++cg) {       // 16 groups of 4 channels
      const int cpair = cg * 2 + lh;        // channel pair index
      const float* p = xbase + (size_t)cg * (2 * CPROW);
      v2f bfa = *(const v2f*)(p);           // {x[c][j],   x[c+1][j]}
      v2f bfb = *(const v2f*)(p + 32);      // {x[c][j+16],x[c+1][j+16]}
      v2f a0 = *(const v2f*)(w0b + (size_t)cpair * (OCH * 2));
      v2f a1 = *(const v2f*)(w1b + (size_t)cpair * (OCH * 2));
      v2f a2 = *(const v2f*)(w2b + (size_t)cpair * (OCH * 2));
      d0a = __builtin_amdgcn_wmma_f32_16x16x4_f32(false, a0, false, bfa,
                                                  (short)0, d0a, false, false);
      d0b = __builtin_amdgcn_wmma_f32_16x16x4_f32(false, a0, false, bfb,
                                                  (short)0, d0b, false, false);
      d1a = __builtin_amdgcn_wmma_f32_16x16x4_f32(false, a1, false, bfa,
                                                  (short)0, d1a, false, false);
      d1b = __builtin_amdgcn_wmma_f32_16x16x4_f32(false, a1, false, bfb,
                                                  (short)0, d1b, false, false);
      d2a = __builtin_amdgcn_wmma_f32_16x16x4_f32(false, a2, false, bfa,
                                                  (short)0, d2a, false, false);
      d2b = __builtin_amdgcn_wmma_f32_16x16x4_f32(false, a2, false, bfb,
                                                  (short)0, d2b, false, false);
    }
    // Fold the per-column masks into the running accumulators.
#pragma unroll
    for (int r = 0; r < 8; ++r) {
      acc0[r] += m0a * d0a[r] + m1a * d1a[r] + m2a * d2a[r];
      acc1[r] += m0b * d0b[r] + m1b * d1b[r] + m2b * d2b[r];
    }
  }

  // ---- store D: VGPR r holds M = r + 8*lh (lanes split), N = jm ----
  float* on = out + (size_t)n * OCH * HH * WW + (size_t)h * WW;
#pragma unroll
  for (int r = 0; r < 8; ++r) {
    int o = o0 + r + 8 * lh;
    on[(size_t)o * HH * WW + j0 + jm]      = acc0[r];
    on[(size_t)o * HH * WW + j0 + 16 + jm] = acc1[r];
  }
}

extern "C" void kernel_launch(void* const* d_in, const int* in_sizes, int n_in,
                              void* d_out, int out_size, void* d_ws, size_t ws_size,
                              hipStream_t stream) {
  const float* x     = (const float*)d_in[0];
  const float* depth = (const float*)d_in[1];
  const float* fx    = (const float*)d_in[2];
  const float* w0    = (const float*)d_in[3];
  const float* w1    = (const float*)d_in[4];
  const float* w2    = (const float*)d_in[5];
  float* out = (float*)d_out;
  float* wt  = (float*)d_ws;   // 3*9*64*64 floats = 442 KB

  prepack_weights<<<(3 * KK * CCH * OCH) / 256, 256, 0, stream>>>(w0, w1, w2, wt);
  conv25d_wmma_kernel<<<8 * HH, 512, 0, stream>>>(x, depth, fx, wt, out);
}